// _System_general_52759378264786
// MI455X (gfx1250) — compile-verified
//
#include <hip/hip_runtime.h>
#include <hip/hip_bf16.h>
#include <math.h>

// ---------------------------------------------------------------------------
// REN forward step for MI455X (gfx1250, wave32)
//   - all GEMMs on V_WMMA_F32_16X16X4_F32 (fp32-exact vs reference)
//   - global->LDS staging via GLOBAL_LOAD_ASYNC_TO_LDS_B128 (ASYNCcnt),
//     double-buffered so DMA overlaps the WMMA stream
//   - E^{-1} via Newton-Schulz (E = I + small perturbation)
//   - sequential tanh column recurrence in LDS (one wave / 32 batch rows)
// ---------------------------------------------------------------------------

#define NB 8192
#define NX 512
#define NU 128
#define NY 128
#define NQ 256

typedef float v2f __attribute__((ext_vector_type(2)));
typedef float v8f __attribute__((ext_vector_type(8)));

#define TILE_M 128
#define TILE_N 128
#define KC 16
#define TSTR 20  // dword stride of one tile row (16 data + 4 pad); 20*4B % 16B == 0

// Issue async copy of a 128x16 tile (rows row0.., cols kc..kc+15 of G[ld]) into
// lds rows of stride TSTR. 4 chunks/row of 16B; 512 chunks over 256 threads
// -> 2 async instructions per thread (= 4/wave for A+B per stage).
__device__ __forceinline__ void async_stage_tile(
    const float* __restrict__ G, int ld, int row0, int kc,
    float* lds_base, int tid)
{
#pragma unroll
    for (int f = tid; f < 512; f += 256) {
        const int r  = f >> 2;
        const int c4 = (f & 3) << 2;
        const float* g = G + (size_t)(row0 + r) * ld + kc + c4;
        const unsigned lds = (unsigned)(uintptr_t)(lds_base + r * TSTR + c4);
        asm volatile("global_load_async_to_lds_b128 %0, %1, off"
                     :: "v"(lds), "v"(g) : "memory");
    }
}

__device__ __forceinline__ void mma_tile(
    const float* __restrict__ As, const float* __restrict__ Bs,
    v8f (&acc)[2][4], int wm, int wn, int l16, int khalf)
{
#pragma unroll
    for (int kk = 0; kk < KC; kk += 4) {
        const int kr = kk + khalf;  // even -> 8B aligned v2f loads
        v2f a[2], b[4];
#pragma unroll
        for (int i = 0; i < 2; ++i)
            a[i] = *(const v2f*)(As + (wm * 32 + i * 16 + l16) * TSTR + kr);
#pragma unroll
        for (int j = 0; j < 4; ++j)
            b[j] = *(const v2f*)(Bs + (wn * 64 + j * 16 + l16) * TSTR + kr);
#pragma unroll
        for (int i = 0; i < 2; ++i)
#pragma unroll
            for (int j = 0; j < 4; ++j)
                acc[i][j] = __builtin_amdgcn_wmma_f32_16x16x4_f32(
                    false, a[i], false, b[j], (short)0, acc[i][j], false, false);
    }
}

// Generic fused GEMM: out[M,N] = sum_p A_p (M x K_p) * B_p + bias[n] (+2I)
//  trans_b=1: W_p is [N x K_p] row-major -> out = A @ W^T  (async pipelined)
//  trans_b=0: W_p is [K_p x N] row-major -> out = A @ W    (sync staging)
// grid = (N/128, M/128), block = 256 threads (8 waves: 4 along M x 2 along N)
__launch_bounds__(256)
__global__ void gemm_wmma_f32(
    const float* __restrict__ A0, const float* __restrict__ W0, int K0,
    const float* __restrict__ A1, const float* __restrict__ W1, int K1,
    const float* __restrict__ A2, const float* __restrict__ W2, int K2,
    int npairs, const float* __restrict__ bias, int add_ident2, int trans_b,
    int N, float* __restrict__ out)
{
    __shared__ __align__(16) float As[2][TILE_M * TSTR];
    __shared__ __align__(16) float Bs[2][TILE_N * TSTR];

    const int tid   = threadIdx.x;
    const int lane  = tid & 31;
    const int wave  = tid >> 5;
    const int wm    = wave & 3;
    const int wn    = wave >> 2;
    const int bm    = blockIdx.y * TILE_M;
    const int bn    = blockIdx.x * TILE_N;
    const int l16   = lane & 15;
    const int khalf = (lane >> 4) << 1;  // +0 lanes 0-15, +2 lanes 16-31

    v8f acc[2][4];
#pragma unroll
    for (int i = 0; i < 2; ++i)
#pragma unroll
        for (int j = 0; j < 4; ++j)
            acc[i][j] = (v8f)(0.0f);

    const float* Ap[3] = {A0, A1, A2};
    const float* Wp[3] = {W0, W1, W2};
    const int    Kp[3] = {K0, K1, K2};
    int tcnt[3];
#pragma unroll
    for (int p = 0; p < 3; ++p) tcnt[p] = (p < npairs) ? (Kp[p] >> 4) : 0;
    const int total = tcnt[0] + tcnt[1] + tcnt[2];

    if (trans_b) {
        // ---- async double-buffered pipeline over flattened tile list ----
        // map t -> (pair p, kc)
        auto issue = [&](int t, int buf) {
            int p, kc;
            if (t < tcnt[0])           { p = 0; kc = t << 4; }
            else if (t < tcnt[0] + tcnt[1]) { p = 1; kc = (t - tcnt[0]) << 4; }
            else                       { p = 2; kc = (t - tcnt[0] - tcnt[1]) << 4; }
            async_stage_tile(Ap[p], Kp[p], bm, kc, &As[buf][0], tid);
            async_stage_tile(Wp[p], Kp[p], bn, kc, &Bs[buf][0], tid);
        };
        issue(0, 0);
        for (int t = 0; t < total; ++t) {
            const int buf = t & 1;
            if (t + 1 < total) {
                issue(t + 1, buf ^ 1);
                // 4 async ops/wave per stage; <=4 outstanding => stage t done
                asm volatile("s_wait_asynccnt 0x4" ::: "memory");
            } else {
                asm volatile("s_wait_asynccnt 0x0" ::: "memory");
            }
            __syncthreads();
            mma_tile(&As[buf][0], &Bs[buf][0], acc, wm, wn, l16, khalf);
            __syncthreads();  // safe to overwrite buf next iteration
        }
    } else {
        // ---- synchronous staging (tiny Newton/solve GEMMs) ----
        for (int p = 0; p < npairs; ++p) {
            const float* __restrict__ A = Ap[p];
            const float* __restrict__ W = Wp[p];
            const int K = Kp[p];
            for (int kc = 0; kc < K; kc += KC) {
#pragma unroll
                for (int f = tid; f < 512; f += 256) {
                    const int r  = f >> 2;
                    const int c4 = (f & 3) << 2;
                    *(float4*)(&As[0][r * TSTR + c4]) =
                        *(const float4*)(A + (size_t)(bm + r) * K + kc + c4);
                }
#pragma unroll
                for (int f = tid; f < 512; f += 256) {
                    const int kk = f >> 5;         // 0..15
                    const int nc = (f & 31) << 2;  // 0..124
                    const float4 v = *(const float4*)(W + (size_t)(kc + kk) * N + bn + nc);
                    Bs[0][(nc + 0) * TSTR + kk] = v.x;
                    Bs[0][(nc + 1) * TSTR + kk] = v.y;
                    Bs[0][(nc + 2) * TSTR + kk] = v.z;
                    Bs[0][(nc + 3) * TSTR + kk] = v.w;
                }
                __syncthreads();
                mma_tile(&As[0][0], &Bs[0][0], acc, wm, wn, l16, khalf);
                __syncthreads();
            }
        }
    }

    // ---- epilogue: D layout: VGPR e -> row = e (+8 for lanes 16-31), col = lane%16
    const int hi8 = (lane >> 4) << 3;
#pragma unroll
    for (int i = 0; i < 2; ++i) {
#pragma unroll
        for (int j = 0; j < 4; ++j) {
            const int col = bn + wn * 64 + j * 16 + l16;
            const float bb = bias ? bias[col] : 0.0f;
#pragma unroll
            for (int e = 0; e < 8; ++e) {
                const int row = bm + wm * 32 + i * 16 + hi8 + e;
                float val = acc[i][j][e] + bb;
                if (add_ident2 && row == col) val += 2.0f;
                out[(size_t)row * N + col] = val;
            }
        }
    }
}

// prep: Eneg = -E ; X0 = 2I - E  (first Newton step from X=I)
__global__ void prep_newton(const float* __restrict__ E,
                            float* __restrict__ Eneg, float* __restrict__ X0)
{
    const int i = blockIdx.x * 256 + threadIdx.x;
    if (i >= NX * NX) return;
    const float e = E[i];
    Eneg[i] = -e;
    const int r = i >> 9, c = i & (NX - 1);
    X0[i] = (r == c ? 2.0f : 0.0f) - e;
}

// Sequential column recurrence:
//   w[:,k] = tanh((base[:,k] + sum_{j<k} w[:,j]*D11[k,j]) / lam[k])
// One wave (32 threads) per 32 batch rows; each lane's w-row is private in LDS.
#define WROWS 32
#define WSTR 257  // odd stride -> bank-conflict-free per-lane row access
__launch_bounds__(32)
__global__ void ren_w_kernel(const float* __restrict__ base,
                             const float* __restrict__ D11,
                             const float* __restrict__ Lambda,
                             float* __restrict__ w_out)
{
    __shared__ float w_s[WROWS * WSTR];
    __shared__ float d11row[NQ];
    __shared__ float lam_s[NQ];

    const int lane = threadIdx.x;
    const int row0 = blockIdx.x * WROWS;
    const int r0   = lane * WSTR;

#pragma unroll
    for (int i = 0; i < NQ / 32; ++i) {
        const int k = lane + 32 * i;
        lam_s[k] = Lambda[(size_t)k * NQ + k];
    }
    for (int r = 0; r < WROWS; ++r) {
#pragma unroll
        for (int i = 0; i < NQ / 32; ++i) {
            const int c = lane + 32 * i;
            w_s[r * WSTR + c] = base[(size_t)(row0 + r) * NQ + c];
        }
    }
    __syncthreads();

    for (int k = 0; k < NQ; ++k) {
#pragma unroll
        for (int i = 0; i < NQ / 32; ++i) {
            const int c = lane + 32 * i;
            d11row[c] = D11[(size_t)k * NQ + c];
        }
        __syncthreads();

        float v = w_s[r0 + k];  // = base[row,k]
        int j = 0;
        for (; j + 3 < k; j += 4) {
            v += w_s[r0 + j + 0] * d11row[j + 0];
            v += w_s[r0 + j + 1] * d11row[j + 1];
            v += w_s[r0 + j + 2] * d11row[j + 2];
            v += w_s[r0 + j + 3] * d11row[j + 3];
        }
        for (; j < k; ++j) v += w_s[r0 + j] * d11row[j];
        v /= lam_s[k];
        w_s[r0 + k] = tanhf(v);
        __syncthreads();  // WAR on d11row
    }

    for (int r = 0; r < WROWS; ++r) {
#pragma unroll
        for (int i = 0; i < NQ / 32; ++i) {
            const int c = lane + 32 * i;
            w_out[(size_t)(row0 + r) * NQ + c] = w_s[r * WSTR + c];
        }
    }
}

extern "C" void kernel_launch(void* const* d_in, const int* in_sizes, int n_in,
                              void* d_out, int out_size, void* d_ws, size_t ws_size,
                              hipStream_t stream) {
    const float* x      = (const float*)d_in[0];   // 8192 x 512
    const float* u      = (const float*)d_in[1];   // 8192 x 128
    const float* C1     = (const float*)d_in[2];   // 256 x 512
    const float* D11    = (const float*)d_in[3];   // 256 x 256
    const float* D12    = (const float*)d_in[4];   // 256 x 128
    const float* Lambda = (const float*)d_in[5];   // 256 x 256
    const float* bv     = (const float*)d_in[6];   // 256
    const float* E      = (const float*)d_in[7];   // 512 x 512
    const float* F      = (const float*)d_in[8];   // 512 x 512
    const float* B1     = (const float*)d_in[9];   // 512 x 256
    const float* B2     = (const float*)d_in[10];  // 512 x 128
    const float* C2     = (const float*)d_in[11];  // 128 x 512
    const float* D21    = (const float*)d_in[12];  // 128 x 256

    float* out_dx = (float*)d_out;                       // 8192 x 512
    float* out_y  = (float*)d_out + (size_t)NB * NX;     // 8192 x 128

    float* ws   = (float*)d_ws;
    float* Eneg = ws;                         // 512*512
    float* Xa   = Eneg + NX * NX;             // 512*512
    float* Xb   = Xa   + NX * NX;             // 512*512
    float* T    = Xb   + NX * NX;             // 512*512
    float* W1   = T    + NX * NX;             // 512*512  Einv@F
    float* W2   = W1   + NX * NX;             // 512*256  Einv@B1
    float* W3   = W2   + NX * NQ;             // 512*128  Einv@B2
    float* base = W3   + NX * NU;             // 8192*256
    float* wbuf = base + (size_t)NB * NQ;     // 8192*256

    const dim3 blk(256);

    // 1) Newton-Schulz setup: Eneg = -E, X0 = 2I - E
    prep_newton<<<(NX * NX + 255) / 256, blk, 0, stream>>>(E, Eneg, Xa);

    // 2) Newton iterations: T = 2I - E@X ; X = X@T   (-> fp32-exact E^{-1})
    float* X  = Xa;
    float* Xn = Xb;
    for (int it = 0; it < 3; ++it) {
        gemm_wmma_f32<<<dim3(NX / TILE_N, NX / TILE_M), blk, 0, stream>>>(
            Eneg, X, NX,  x, x, 0,  x, x, 0,
            1, nullptr, /*add2I=*/1, /*trans_b=*/0, NX, T);
        gemm_wmma_f32<<<dim3(NX / TILE_N, NX / TILE_M), blk, 0, stream>>>(
            X, T, NX,  x, x, 0,  x, x, 0,
            1, nullptr, 0, /*trans_b=*/0, NX, Xn);
        float* tmp = X; X = Xn; Xn = tmp;
    }

    // 3) W1 = Einv@F, W2 = Einv@B1, W3 = Einv@B2
    gemm_wmma_f32<<<dim3(NX / TILE_N, NX / TILE_M), blk, 0, stream>>>(
        X, F, NX, x, x, 0, x, x, 0, 1, nullptr, 0, 0, NX, W1);
    gemm_wmma_f32<<<dim3(NQ / TILE_N, NX / TILE_M), blk, 0, stream>>>(
        X, B1, NX, x, x, 0, x, x, 0, 1, nullptr, 0, 0, NQ, W2);
    gemm_wmma_f32<<<dim3(NU / TILE_N, NX / TILE_M), blk, 0, stream>>>(
        X, B2, NX, x, x, 0, x, x, 0, 1, nullptr, 0, 0, NU, W3);

    // 4) base = x@C1^T + u@D12^T + bv
    gemm_wmma_f32<<<dim3(NQ / TILE_N, NB / TILE_M), blk, 0, stream>>>(
        x, C1, NX,  u, D12, NU,  x, x, 0,
        2, bv, 0, /*trans_b=*/1, NQ, base);

    // 5) sequential tanh recurrence -> w
    ren_w_kernel<<<NB / WROWS, dim3(32), 0, stream>>>(base, D11, Lambda, wbuf);

    // 6) dx = x@W1^T + w@W2^T + u@W3^T
    gemm_wmma_f32<<<dim3(NX / TILE_N, NB / TILE_M), blk, 0, stream>>>(
        x, W1, NX,  wbuf, W2, NQ,  u, W3, NU,
        3, nullptr, 0, /*trans_b=*/1, NX, out_dx);

    // 7) y = x@C2^T + w@D21^T
    gemm_wmma_f32<<<dim3(NY / TILE_N, NB / TILE_M), blk, 0, stream>>>(
        x, C2, NX,  wbuf, D21, NQ,  x, x, 0,
        2, nullptr, 0, /*trans_b=*/1, NY, out_y);
}